// pnablock_70600672411874
// MI455X (gfx1250) — compile-verified
//
#include <hip/hip_runtime.h>
#include <math.h>

typedef float v2f __attribute__((ext_vector_type(2)));
typedef float v8f __attribute__((ext_vector_type(8)));

#define D 128
#define DELTA_INV 0.4f        /* 1/2.5 */
#define BN_EPS 1e-5f
#define NEG_SLOPE 0.01f

__device__ __forceinline__ v8f wmma4(v2f a, v2f b, v8f c) {
  // V_WMMA_F32_16X16X4_F32 : D = A(16x4) * B(4x16) + C(16x16), fp32
  return __builtin_amdgcn_wmma_f32_16x16x4_f32(false, a, false, b, (short)0, c,
                                               false, false);
}

// float atomic max via sign-split integer atomics (target init must be -inf)
__device__ __forceinline__ void atomicMaxF(float* addr, float v) {
  if (v >= 0.0f) atomicMax((int*)addr, __float_as_int(v));
  else           atomicMin((unsigned int*)addr, __float_as_uint(v));
}

// ---------------- kernel 0: init scratch ----------------
__global__ void k_init(float* __restrict__ sum, float* __restrict__ mx,
                       float* __restrict__ deg, float* __restrict__ colsum,
                       float* __restrict__ colsq, int N) {
  size_t i = (size_t)blockIdx.x * blockDim.x + threadIdx.x;
  size_t nd = (size_t)N * D;
  if (i < nd) { sum[i] = 0.0f; mx[i] = -__builtin_inff(); }
  if (i < (size_t)N) deg[i] = 0.0f;
  if (i < D) { colsum[i] = 0.0f; colsq[i] = 0.0f; }
}

// ---------------- kernel 1: in-degree (with self loops) ----------------
__global__ void k_deg(const int* __restrict__ dst, float* __restrict__ deg,
                      int E, int ET) {
  int e = blockIdx.x * blockDim.x + threadIdx.x;
  if (e < ET) {
    int dd = (e < E) ? dst[e] : (e - E);
    atomicAdd(&deg[dd], 1.0f);
  }
}

// ------------- kernel 2: edge message GEMM + fused mean/max aggregation -----
// wave w of each block owns n-tile w (16 output cols); M_w strip lives in regs.
__global__ void __launch_bounds__(256) k_edge(
    const float* __restrict__ ndata, const float* __restrict__ Mw,
    const float* __restrict__ Mb, const int* __restrict__ src,
    const int* __restrict__ dst, float* __restrict__ sum,
    float* __restrict__ mx, int E, int ET, int numTiles) {
  const int lane = threadIdx.x & 31;
  const int w    = threadIdx.x >> 5;
  const int j    = lane & 15;       // row-within-tile / col-within-ntile
  const int lh   = lane >> 4;       // lane half
  const int n0   = w * 16;
  const int col  = n0 + j;
  const int kb   = 2 * lh;

  // Register-resident B strip: B[k].x = Mw[(4k+kb)][col], .y = Mw[(4k+kb+1)][col]
  v2f B[64];
#pragma unroll
  for (int k = 0; k < 64; ++k) {
    int kk = 4 * k + kb;
    B[k].x = Mw[(size_t)kk * D + col];
    B[k].y = Mw[(size_t)(kk + 1) * D + col];
  }
  const float bias = Mb[col];

  for (int t = blockIdx.x; t < numTiles; t += gridDim.x) {
    const int base = t * 16;
    int et  = base + j;
    int etc = (et < ET) ? et : ET - 1;
    int srow = (etc < E) ? src[etc] : (etc - E);
    int drow = (etc < E) ? dst[etc] : (etc - E);
    v8f acc = {};
#pragma unroll
    for (int k = 0; k < 64; ++k) {
      int ko  = 4 * k + kb;                // K index pair start
      int row = (ko < 128) ? srow : drow;  // concat [h_src | h_dst]
      int c   = ko & 127;
      v2f a = *(const v2f*)(ndata + (size_t)row * D + c);
      acc = wmma4(a, B[k], acc);
    }
#pragma unroll
    for (int v = 0; v < 8; ++v) {
      int mrow = v + 8 * lh;
      int e2   = base + mrow;
      if (e2 < ET) {
        int dd = (e2 < E) ? dst[e2] : (e2 - E);
        float val = acc[v] + bias;
        atomicAdd(&sum[(size_t)dd * D + col], val);
        atomicMaxF(&mx[(size_t)dd * D + col], val);
      }
    }
  }
}

// ---------------- kernel 3a: mean in place; 3b: amp ----------------
__global__ void k_mean(float* __restrict__ sum, const float* __restrict__ deg,
                       size_t nd) {
  size_t i = (size_t)blockIdx.x * blockDim.x + threadIdx.x;
  if (i < nd) sum[i] /= deg[i >> 7];
}
__global__ void k_amp(const float* __restrict__ deg, float* __restrict__ amp,
                      int N) {
  int i = blockIdx.x * blockDim.x + threadIdx.x;
  if (i < N) amp[i] = logf(deg[i] + 1.0f) * DELTA_INV;
}

// ------ kernel 4: U GEMM (K=896 synthesized concat) + BN partial sums ------
__global__ void __launch_bounds__(256) k_unet(
    const float* __restrict__ ndata, const float* __restrict__ mean,
    const float* __restrict__ mx, const float* __restrict__ amp,
    const float* __restrict__ Uw, const float* __restrict__ Ub,
    float* __restrict__ h, float* __restrict__ colsum,
    float* __restrict__ colsq, int N, int rowTiles) {
  __shared__ float Bs[896 * 16];  // 56 KB U_w strip for this n-tile
  const int n0 = blockIdx.y * 16;
  for (int i = threadIdx.x; i < 896 * 16; i += 256)
    Bs[i] = Uw[(size_t)(i >> 4) * D + n0 + (i & 15)];
  __syncthreads();

  const int lane = threadIdx.x & 31;
  const int w    = threadIdx.x >> 5;
  const int j    = lane & 15;
  const int lh   = lane >> 4;
  const int col  = n0 + j;
  const int kb   = 2 * lh;
  int rt = blockIdx.x * 8 + w;
  if (rt >= rowTiles) return;
  const int base = rt * 16;
  int node = base + j;
  int nc   = (node < N) ? node : N - 1;
  const float ampv = amp[nc];

  v8f acc = {};
#pragma unroll 4
  for (int k = 0; k < 224; ++k) {
    int ko  = 4 * k + kb;
    int seg = ko >> 7;      // wave-uniform: K-pairs never straddle 128 boundary
    int c   = ko & 127;
    const float* bp; bool sc = false;
    switch (seg) {
      case 0: bp = ndata; break;
      case 1: case 3: bp = mean; break;
      case 2: bp = mx; break;
      case 4: case 6: bp = mean; sc = true; break;
      default: bp = mx; sc = true; break;
    }
    v2f a = *(const v2f*)(bp + (size_t)nc * D + c);
    if (sc) { a.x *= ampv; a.y *= ampv; }
    v2f b;
    b.x = Bs[ko * 16 + j];
    b.y = Bs[(ko + 1) * 16 + j];
    acc = wmma4(a, b, acc);
  }
  const float bias = Ub[col];
  float s1 = 0.0f, s2 = 0.0f;
#pragma unroll
  for (int v = 0; v < 8; ++v) {
    int nrow = base + v + 8 * lh;
    if (nrow < N) {
      float val = acc[v] + bias;
      h[(size_t)nrow * D + col] = val;
      s1 += val; s2 += val * val;
    }
  }
  atomicAdd(&colsum[col], s1);
  atomicAdd(&colsq[col], s2);
}

// ---------------- kernel 5: BN finalize -> per-column affine ----------------
__global__ void k_bn(const float* __restrict__ colsum,
                     const float* __restrict__ colsq,
                     const float* __restrict__ gamma,
                     const float* __restrict__ beta,
                     float* __restrict__ bnsc, float* __restrict__ bnsh, int N) {
  int c = threadIdx.x;
  float inv = 1.0f / (float)N;
  float mu  = colsum[c] * inv;
  float var = colsq[c] * inv - mu * mu;
  var = var > 0.0f ? var : 0.0f;
  float s = gamma[c] * rsqrtf(var + BN_EPS);
  bnsc[c] = s;
  bnsh[c] = beta[c] - mu * s;
}

// -------- kernel 6: mix GEMM with BN folded into A-load + LeakyReLU --------
__global__ void __launch_bounds__(256) k_mix(
    const float* __restrict__ h, const float* __restrict__ bnsc,
    const float* __restrict__ bnsh, const float* __restrict__ Ww,
    const float* __restrict__ Wb, float* __restrict__ out, int N,
    int rowTiles) {
  __shared__ float Bs[128 * 16];
  __shared__ float ssc[128], ssh[128];
  const int n0 = blockIdx.y * 16;
  for (int i = threadIdx.x; i < 128 * 16; i += 256)
    Bs[i] = Ww[(size_t)(i >> 4) * D + n0 + (i & 15)];
  if (threadIdx.x < 128) { ssc[threadIdx.x] = bnsc[threadIdx.x];
                           ssh[threadIdx.x] = bnsh[threadIdx.x]; }
  __syncthreads();

  const int lane = threadIdx.x & 31;
  const int w    = threadIdx.x >> 5;
  const int j    = lane & 15;
  const int lh   = lane >> 4;
  const int col  = n0 + j;
  const int kb   = 2 * lh;
  int rt = blockIdx.x * 8 + w;
  if (rt >= rowTiles) return;
  const int base = rt * 16;
  int node = base + j;
  int nc   = (node < N) ? node : N - 1;

  v8f acc = {};
#pragma unroll
  for (int k = 0; k < 32; ++k) {
    int ko = 4 * k + kb;
    v2f a = *(const v2f*)(h + (size_t)nc * D + ko);
    a.x = a.x * ssc[ko] + ssh[ko];
    a.y = a.y * ssc[ko + 1] + ssh[ko + 1];
    v2f b;
    b.x = Bs[ko * 16 + j];
    b.y = Bs[(ko + 1) * 16 + j];
    acc = wmma4(a, b, acc);
  }
  const float bias = Wb[col];
#pragma unroll
  for (int v = 0; v < 8; ++v) {
    int nrow = base + v + 8 * lh;
    if (nrow < N) {
      float val = acc[v] + bias;
      out[(size_t)nrow * D + col] = val > 0.0f ? val : NEG_SLOPE * val;
    }
  }
}

extern "C" void kernel_launch(void* const* d_in, const int* in_sizes, int n_in,
                              void* d_out, int out_size, void* d_ws,
                              size_t ws_size, hipStream_t stream) {
  const float* ndata = (const float*)d_in[0];
  const float* Mw    = (const float*)d_in[1];
  const float* Mb    = (const float*)d_in[2];
  const float* Uw    = (const float*)d_in[3];
  const float* Ub    = (const float*)d_in[4];
  const float* gam   = (const float*)d_in[5];
  const float* bet   = (const float*)d_in[6];
  const float* mixw  = (const float*)d_in[7];
  const float* mixb  = (const float*)d_in[8];
  const int*   src   = (const int*)d_in[9];
  const int*   dst   = (const int*)d_in[10];

  const int N  = in_sizes[0] / D;
  const int E  = in_sizes[9];
  const int ET = E + N;
  const size_t nd = (size_t)N * D;

  float* ws     = (float*)d_ws;
  float* sum    = ws;               // (N,128)  -> becomes mean in place
  float* mx     = ws + nd;          // (N,128)
  float* hbuf   = ws + 2 * nd;      // (N,128)
  float* deg    = ws + 3 * nd;      // (N,)
  float* amp    = deg + N;          // (N,)
  float* colsum = amp + N;          // (128,)
  float* colsq  = colsum + D;       // (128,)
  float* bnsc   = colsq + D;        // (128,)
  float* bnsh   = bnsc + D;         // (128,)

  const int numTiles = (ET + 15) / 16;
  const int rowTiles = (N + 15) / 16;

  k_init<<<(int)((nd + 255) / 256), 256, 0, stream>>>(sum, mx, deg, colsum,
                                                      colsq, N);
  k_deg<<<(ET + 255) / 256, 256, 0, stream>>>(dst, deg, E, ET);

  int g1 = numTiles < 2048 ? numTiles : 2048;
  k_edge<<<g1, 256, 0, stream>>>(ndata, Mw, Mb, src, dst, sum, mx, E, ET,
                                 numTiles);

  k_mean<<<(int)((nd + 255) / 256), 256, 0, stream>>>(sum, deg, nd);
  k_amp<<<(N + 255) / 256, 256, 0, stream>>>(deg, amp, N);

  dim3 g2((rowTiles + 7) / 8, 8);
  k_unet<<<g2, 256, 0, stream>>>(ndata, sum, mx, amp, Uw, Ub, hbuf, colsum,
                                 colsq, N, rowTiles);

  k_bn<<<1, 128, 0, stream>>>(colsum, colsq, gam, bet, bnsc, bnsh, N);

  k_mix<<<g2, 256, 0, stream>>>(hbuf, bnsc, bnsh, mixw, mixb, (float*)d_out, N,
                                rowTiles);
}